// PCT_Tokenizer_42142219108404
// MI455X (gfx1250) — compile-verified
//
#include <hip/hip_runtime.h>
#include <hip/hip_bf16.h>
#include <math.h>

// ---------------------------------------------------------------------------
// PCT tokenizer forward for MI455X (gfx1250, wave32).
// Big fp32 GEMMs (encoder channel-mix, feat proj, VQ scores) run on the matrix
// pipe via v_wmma_f32_16x16x4_f32 (precision-exact vs fp32 reference).
// Small-K contractions (J=17, TNUM=34, DHID=32) stay VALU (bandwidth bound).
// GEMM: 128x64 macro-tile, 8 waves (one 16-row slice each, 4 accumulators),
// float4 global staging, W tile stored transposed in LDS so every WMMA
// fragment is a single 8B-aligned ds_load_b64. N assumed multiple of 4.
// ---------------------------------------------------------------------------

typedef __attribute__((ext_vector_type(2))) float v2f;
typedef __attribute__((ext_vector_type(4))) float v4f;
typedef __attribute__((ext_vector_type(8))) float v8f;

#define BN 4096
#define JN 17
#define HIDD 512
#define TINT 64
#define TNUM 34
#define TDIM 512
#define KCLS 2048
#define DHID 32
#define DINT 64

__device__ __forceinline__ float gelu_exact(float x) {
  return 0.5f * x * (1.0f + erff(x * 0.70710678118654752440f));
}

// ---------------------------------------------------------------- zero loss
__global__ void k_zero(float* p) {
  if (threadIdx.x == 0 && blockIdx.x == 0) p[0] = 0.0f;
}

// ------------------------------------------------- encoder start + vis mask
// O[row,d] = vis[row] ? joints[row]@sw + sb : inv_token[d]
__global__ __launch_bounds__(256) void k_start(
    const float* __restrict__ joints, const int* __restrict__ vis,
    const float* __restrict__ sw, const float* __restrict__ sb,
    const float* __restrict__ invtok, float* __restrict__ O) {
  const size_t id = (size_t)blockIdx.x * 256 + threadIdx.x;  // B*J*512
  const size_t row = id >> 9;
  const int d = (int)(id & 511);
  const float* j = joints + row * 3;
  float x = j[0] * sw[d] + j[1] * sw[512 + d] + j[2] * sw[1024 + d] + sb[d];
  O[id] = vis[row] ? x : invtok[d];
}

// ------------------------------------------------------ LayerNorm, dim 512
// one wave per row, 8 rows per 256-thread block
__global__ __launch_bounds__(256) void k_ln512(
    const float* __restrict__ X, const float* __restrict__ g,
    const float* __restrict__ b, float* __restrict__ O) {
  const int row = blockIdx.x * 8 + (threadIdx.x >> 5);
  const int lane = threadIdx.x & 31;
  const float* x = X + (size_t)row * HIDD;
  float v[16];
  float s = 0.f;
#pragma unroll
  for (int i = 0; i < 16; ++i) { v[i] = x[lane + 32 * i]; s += v[i]; }
#pragma unroll
  for (int o = 16; o > 0; o >>= 1) s += __shfl_xor(s, o, 32);
  const float mean = s * (1.f / 512.f);
  float q = 0.f;
#pragma unroll
  for (int i = 0; i < 16; ++i) { float d = v[i] - mean; q += d * d; }
#pragma unroll
  for (int o = 16; o > 0; o >>= 1) q += __shfl_xor(q, o, 32);
  const float rstd = rsqrtf(q * (1.f / 512.f) + 1e-5f);
  float* out = O + (size_t)row * HIDD;
#pragma unroll
  for (int i = 0; i < 16; ++i) {
    const int d = lane + 32 * i;
    out[d] = (v[i] - mean) * rstd * g[d] + b[d];
  }
}

// ------------------------------------------- encoder token-mix + residual
// thread per (b,d): XY[b,j,d] = X[b,j,d] + MLP_over_j(Y[b,:,d])
__global__ __launch_bounds__(256) void k_tokmix(
    const float* __restrict__ X, const float* __restrict__ Y,
    const float* __restrict__ tw1, const float* __restrict__ tb1,
    const float* __restrict__ tw2, const float* __restrict__ tb2,
    float* __restrict__ XY) {
  __shared__ float w1s[JN * TINT], w2s[TINT * JN];
  __shared__ float b1s[TINT], b2s[JN];
  for (int i = threadIdx.x; i < JN * TINT; i += 256) w1s[i] = tw1[i];
  for (int i = threadIdx.x; i < TINT * JN; i += 256) w2s[i] = tw2[i];
  if (threadIdx.x < TINT) b1s[threadIdx.x] = tb1[threadIdx.x];
  if (threadIdx.x < JN) b2s[threadIdx.x] = tb2[threadIdx.x];
  __syncthreads();
  const size_t id = (size_t)blockIdx.x * 256 + threadIdx.x;  // B*512
  const size_t b = id >> 9;
  const int d = (int)(id & 511);
  float y[JN];
#pragma unroll
  for (int j = 0; j < JN; ++j) y[j] = Y[(b * JN + j) * 512 + d];
  float t1[TINT];
  for (int t = 0; t < TINT; ++t) {
    float s = b1s[t];
#pragma unroll
    for (int j = 0; j < JN; ++j) s += y[j] * w1s[j * TINT + t];
    t1[t] = gelu_exact(s);
  }
#pragma unroll
  for (int j = 0; j < JN; ++j) {
    float s = b2s[j];
    for (int t = 0; t < TINT; ++t) s += t1[t] * w2s[t * JN + j];
    const size_t o = (b * JN + j) * 512 + d;
    XY[o] = X[o] + s;
  }
}

// --------------------------------------------------------- f32 WMMA GEMM
// C[M,N] = epilogue(A[M,K] @ W[K,N] + bias [, +resid])
// Macro tile 128x64; wave w owns rows [w*16, w*16+16) x all 64 cols
// (4 v8f accumulators, A fragment reused 4x). M % 128 == 0, N % 4 == 0.
__global__ __launch_bounds__(256) void k_gemm_f32(
    const float* __restrict__ A, const float* __restrict__ W,
    const float* __restrict__ bias, const float* __restrict__ resid,
    float* __restrict__ C, int M, int N, int K, int fuse_gelu) {
  __shared__ float As[128][20];   // stride 20: 16B-aligned rows, no conflicts
  __shared__ float WsT[64][22];   // transposed [n][k]; stride 22: 8B aligned
  const int tid = threadIdx.x;
  const int wave = tid >> 5;  // 0..7 -> m slice
  const int lane = tid & 31;
  const int m0 = blockIdx.x * 128;
  const int n0 = blockIdx.y * 64;

  v8f acc[4] = {{}, {}, {}, {}};

  const int am = wave * 16 + (lane & 15);
  const int kb = (lane >> 4) * 2;  // K sub-pair per half-wave
  const int bn = lane & 15;

  const int wr = tid >> 4;          // W stage: k row 0..15
  const int wc4 = (tid & 15) * 4;   // W stage: n chunk
  const bool w_in = (n0 + wc4) < N; // N%4==0 && wc4%4==0 => whole f4 in range

  for (int k0 = 0; k0 < K; k0 += 16) {
    // stage A: 128 rows x 16 k = 512 float4, 2 per thread (b128 loads/stores)
#pragma unroll
    for (int i = 0; i < 2; ++i) {
      const int t = tid + i * 256;
      const int r = t >> 2, c4 = (t & 3) * 4;
      *(v4f*)&As[r][c4] = *(const v4f*)&A[(size_t)(m0 + r) * K + (k0 + c4)];
    }
    // stage W transposed: one predicated float4 per thread, scatter to [n][k]
    {
      v4f w = {};
      if (w_in) w = *(const v4f*)&W[(size_t)(k0 + wr) * N + (n0 + wc4)];
      WsT[wc4 + 0][wr] = w.x;
      WsT[wc4 + 1][wr] = w.y;
      WsT[wc4 + 2][wr] = w.z;
      WsT[wc4 + 3][wr] = w.w;
    }
    __syncthreads();
#pragma unroll
    for (int kk = 0; kk < 16; kk += 4) {
      const v2f a = *(const v2f*)&As[am][kk + kb];  // single ds_load_b64
#pragma unroll
      for (int nt = 0; nt < 4; ++nt) {
        const v2f b = *(const v2f*)&WsT[nt * 16 + bn][kk + kb];
        acc[nt] = __builtin_amdgcn_wmma_f32_16x16x4_f32(
            false, a, false, b, (short)0, acc[nt], false, false);
      }
    }
    __syncthreads();
  }

  const int mb = m0 + wave * 16 + 8 * (lane >> 4);
#pragma unroll
  for (int nt = 0; nt < 4; ++nt) {
    const int n = n0 + nt * 16 + (lane & 15);
    if (n < N) {
      const float bv = bias[n];
#pragma unroll
      for (int r = 0; r < 8; ++r) {
        const int m = mb + r;
        float v = acc[nt][r] + bv;
        if (fuse_gelu) v = gelu_exact(v);
        if (resid) v += resid[(size_t)m * N + n];
        C[(size_t)m * N + n] = v;
      }
    }
  }
}

// ---------------------------------------------- token projection (K=17)
// T[b,t,d] = sum_j E[b,j,d]*tokw[j,t] + tokb[t]
__global__ __launch_bounds__(256) void k_tokproj(
    const float* __restrict__ E, const float* __restrict__ tw,
    const float* __restrict__ tb, float* __restrict__ T) {
  __shared__ float ws[JN * TNUM];
  __shared__ float bs[TNUM];
  for (int i = threadIdx.x; i < JN * TNUM; i += 256) ws[i] = tw[i];
  if (threadIdx.x < TNUM) bs[threadIdx.x] = tb[threadIdx.x];
  __syncthreads();
  const size_t id = (size_t)blockIdx.x * 256 + threadIdx.x;  // B*512
  const size_t b = id >> 9;
  const int d = (int)(id & 511);
  float e[JN];
#pragma unroll
  for (int j = 0; j < JN; ++j) e[j] = E[(b * JN + j) * 512 + d];
  for (int t = 0; t < TNUM; ++t) {
    float s = bs[t];
#pragma unroll
    for (int j = 0; j < JN; ++j) s += e[j] * ws[j * TNUM + t];
    T[(b * TNUM + t) * 512 + d] = s;
  }
}

// ----------------------------------------------------- codebook row norms
__global__ __launch_bounds__(256) void k_rownorm(const float* __restrict__ X,
                                                 float* __restrict__ nrm) {
  const int row = blockIdx.x * 8 + (threadIdx.x >> 5);
  const int lane = threadIdx.x & 31;
  const float* x = X + (size_t)row * TDIM;
  float s = 0.f;
  for (int i = lane; i < TDIM; i += 32) { float v = x[i]; s += v * v; }
#pragma unroll
  for (int o = 16; o > 0; o >>= 1) s += __shfl_xor(s, o, 32);
  if (lane == 0) nrm[row] = s;
}

// ------------------------------------- fused VQ: WMMA scores + argmin + loss
// one workgroup = 16 ef rows; 4 waves each scan 512 codebook entries,
// full-K f32 WMMA accumulation, d = |ef|^2+|cb|^2-2*dot, track min/argmin.
// d_min == |ef-quant|^2, so loss accumulates directly from the minima.
__global__ __launch_bounds__(128) void k_vq(
    const float* __restrict__ EF, const float* __restrict__ CB,
    const float* __restrict__ cbn, int* __restrict__ idxo,
    float* __restrict__ loss, float inv_count) {
  __shared__ float efs[16][516];  // stride 516 -> distinct banks, 8B aligned
  __shared__ float efn[16];
  __shared__ float rv[4][16];
  __shared__ int ri[4][16];
  const int m0 = blockIdx.x * 16;
  for (int i = threadIdx.x; i < 16 * 512; i += 128)
    efs[i >> 9][i & 511] = EF[(size_t)(m0 + (i >> 9)) * 512 + (i & 511)];
  __syncthreads();
  {  // row norms: 8 threads per row, width-8 shuffle reduce
    const int rrow = threadIdx.x >> 3;
    const int sub = threadIdx.x & 7;
    float s = 0.f;
    for (int k = sub; k < 512; k += 8) { float v = efs[rrow][k]; s += v * v; }
#pragma unroll
    for (int o = 4; o > 0; o >>= 1) s += __shfl_xor(s, o, 8);
    if (sub == 0) efn[rrow] = s;
  }
  __syncthreads();
  const int wave = threadIdx.x >> 5;
  const int lane = threadIdx.x & 31;
  const int am = lane & 15;
  const int kb = (lane >> 4) * 2;
  float minv[8];
  int mini[8];
#pragma unroll
  for (int r = 0; r < 8; ++r) { minv[r] = 3.4e38f; mini[r] = 0; }
  const int nbeg = wave * 512;
  for (int nt = nbeg; nt < nbeg + 512; nt += 16) {
    const int n = nt + (lane & 15);
    const float* cbp = CB + (size_t)n * 512;
    v8f acc = {};
    for (int k = 0; k < 512; k += 4) {
      const v2f a = *(const v2f*)&efs[am][k + kb];
      const v2f b = *(const v2f*)(cbp + k + kb);  // 8B global load
      acc = __builtin_amdgcn_wmma_f32_16x16x4_f32(false, a, false, b, (short)0,
                                                  acc, false, false);
    }
    const float cn = cbn[n];
#pragma unroll
    for (int r = 0; r < 8; ++r) {
      const int mm = r + 8 * (lane >> 4);
      const float d = efn[mm] + cn - 2.f * acc[r];
      if (d < minv[r]) { minv[r] = d; mini[r] = n; }
    }
  }
#pragma unroll
  for (int r = 0; r < 8; ++r) {
    float v = minv[r];
    int ii = mini[r];
    for (int o = 8; o > 0; o >>= 1) {  // width-16: reduce lanes sharing m
      const float ov = __shfl_xor(v, o, 16);
      const int oi = __shfl_xor(ii, o, 16);
      if (ov < v) { v = ov; ii = oi; }
    }
    if ((lane & 15) == 0) {
      const int m = r + 8 * (lane >> 4);
      rv[wave][m] = v;
      ri[wave][m] = ii;
    }
  }
  __syncthreads();
  if (threadIdx.x < 16) {
    float bv = rv[0][threadIdx.x];
    int bi = ri[0][threadIdx.x];
    for (int w = 1; w < 4; ++w)
      if (rv[w][threadIdx.x] < bv) { bv = rv[w][threadIdx.x]; bi = ri[w][threadIdx.x]; }
    idxo[m0 + threadIdx.x] = bi;
    atomicAdd(loss, fmaxf(bv, 0.f) * inv_count);
  }
}

// ------------------------- decoder token mix on quantized features (K=34)
// P[b,j,d] = sum_t CB[idx[b,t],d]*dtokw[t,j] + dtokb[j]
__global__ __launch_bounds__(256) void k_dectok(
    const int* __restrict__ idx, const float* __restrict__ CB,
    const float* __restrict__ dw, const float* __restrict__ db,
    float* __restrict__ P) {
  __shared__ float ws[TNUM * JN];
  __shared__ float bs[JN];
  for (int i = threadIdx.x; i < TNUM * JN; i += 256) ws[i] = dw[i];
  if (threadIdx.x < JN) bs[threadIdx.x] = db[threadIdx.x];
  __syncthreads();
  const size_t id = (size_t)blockIdx.x * 256 + threadIdx.x;  // B*512
  const size_t b = id >> 9;
  const int d = (int)(id & 511);
  float q[TNUM];
  for (int t = 0; t < TNUM; ++t)
    q[t] = CB[(size_t)idx[b * TNUM + t] * 512 + d];
#pragma unroll
  for (int j = 0; j < JN; ++j) {
    float s = bs[j];
    for (int t = 0; t < TNUM; ++t) s += q[t] * ws[t * JN + j];
    P[(b * JN + j) * 512 + d] = s;
  }
}

// ------------------ fused decoder mixer + final LN + 32->3 reconstruction
// one block per batch element, 544 = 17*32 threads, all state in LDS.
__global__ __launch_bounds__(544) void k_dec(
    const float* __restrict__ Y0, const float* __restrict__ ln1g,
    const float* __restrict__ ln1b, const float* __restrict__ tw1,
    const float* __restrict__ tb1, const float* __restrict__ tw2,
    const float* __restrict__ tb2, const float* __restrict__ ln2g,
    const float* __restrict__ ln2b, const float* __restrict__ cw1,
    const float* __restrict__ cb1, const float* __restrict__ cw2,
    const float* __restrict__ cb2, const float* __restrict__ flg,
    const float* __restrict__ flb, const float* __restrict__ rw,
    const float* __restrict__ rb, float* __restrict__ out) {
  __shared__ float xs[544], ys[544], xys[544], zs[544], fl[544];
  __shared__ float t1s[DHID * DINT];  // 32x64
  __shared__ float hs[JN * DINT];     // 17x64
  const int b = blockIdx.x;
  const int tid = threadIdx.x;
  xs[tid] = Y0[(size_t)b * 544 + tid];
  __syncthreads();
  if (tid < JN) {  // LN1 per joint row (32 dims)
    const float* xr = xs + tid * DHID;
    float m = 0.f;
    for (int d = 0; d < DHID; ++d) m += xr[d];
    m *= (1.f / DHID);
    float q = 0.f;
    for (int d = 0; d < DHID; ++d) { float dd = xr[d] - m; q += dd * dd; }
    const float rs = rsqrtf(q * (1.f / DHID) + 1e-5f);
    for (int d = 0; d < DHID; ++d)
      ys[tid * DHID + d] = (xr[d] - m) * rs * ln1g[d] + ln1b[d];
  }
  __syncthreads();
  for (int i = tid; i < DHID * DINT; i += 544) {  // token mix up-proj
    const int d = i >> 6, t = i & 63;
    float s = tb1[t];
    for (int j = 0; j < JN; ++j) s += ys[j * DHID + d] * tw1[j * DINT + t];
    t1s[i] = gelu_exact(s);
  }
  __syncthreads();
  {
    const int j = tid >> 5, d = tid & 31;
    float s = tb2[j];
    for (int t = 0; t < DINT; ++t) s += t1s[d * DINT + t] * tw2[t * JN + j];
    xys[tid] = xs[tid] + s;
  }
  __syncthreads();
  if (tid < JN) {  // LN2
    const float* xr = xys + tid * DHID;
    float m = 0.f;
    for (int d = 0; d < DHID; ++d) m += xr[d];
    m *= (1.f / DHID);
    float q = 0.f;
    for (int d = 0; d < DHID; ++d) { float dd = xr[d] - m; q += dd * dd; }
    const float rs = rsqrtf(q * (1.f / DHID) + 1e-5f);
    for (int d = 0; d < DHID; ++d)
      zs[tid * DHID + d] = (xr[d] - m) * rs * ln2g[d] + ln2b[d];
  }
  __syncthreads();
  for (int i = tid; i < JN * DINT; i += 544) {  // channel mix up-proj
    const int j = i >> 6, t = i & 63;
    float s = cb1[t];
    for (int d = 0; d < DHID; ++d) s += zs[j * DHID + d] * cw1[d * DINT + t];
    hs[i] = gelu_exact(s);
  }
  __syncthreads();
  {
    const int j = tid >> 5, d = tid & 31;
    float s = cb2[d];
    for (int t = 0; t < DINT; ++t) s += hs[j * DINT + t] * cw2[t * DHID + d];
    zs[tid] = xys[tid] + s;  // mixer output (safe: all zs reads done)
  }
  __syncthreads();
  if (tid < JN) {  // final LN
    const float* xr = zs + tid * DHID;
    float m = 0.f;
    for (int d = 0; d < DHID; ++d) m += xr[d];
    m *= (1.f / DHID);
    float q = 0.f;
    for (int d = 0; d < DHID; ++d) { float dd = xr[d] - m; q += dd * dd; }
    const float rs = rsqrtf(q * (1.f / DHID) + 1e-5f);
    for (int d = 0; d < DHID; ++d)
      fl[tid * DHID + d] = (xr[d] - m) * rs * flg[d] + flb[d];
  }
  __syncthreads();
  if (tid < JN * 3) {  // reconstruction 32 -> 3
    const int j = tid / 3, c = tid % 3;
    float s = rb[c];
    for (int d = 0; d < DHID; ++d) s += fl[j * DHID + d] * rw[d * 3 + c];
    out[(size_t)b * (JN * 3) + tid] = s;
  }
}

// ---------------------------------------------------------------------------
extern "C" void kernel_launch(void* const* d_in, const int* in_sizes, int n_in,
                              void* d_out, int out_size, void* d_ws,
                              size_t ws_size, hipStream_t stream) {
  (void)in_sizes; (void)n_in; (void)out_size; (void)ws_size;
  auto F = [&](int i) { return (const float*)d_in[i]; };
  // d_in order: joints, vis_mask, start_w, start_b, inv_token,
  //   enc[0..3]{ln1_g,ln1_b,tw1,tb1,tw2,tb2,ln2_g,ln2_b,cw1,cb1,cw2,cb2},
  //   enc_ln_g, enc_ln_b, tok_w, tok_b, feat_w, feat_b, codebook,
  //   dtok_w, dtok_b, dstart_w, dstart_b, dec[0]{...12}, dec_ln_g, dec_ln_b,
  //   rec_w, rec_b
  const float* joints = F(0);
  const int* vis = (const int*)d_in[1];

  float* ws = (float*)d_ws;
  size_t off = 0;
  auto alloc = [&](size_t n) { float* r = ws + off; off += n; return r; };
  const size_t NX = (size_t)BN * JN * HIDD;    // 35.6M floats
  const size_t NT = (size_t)BN * TNUM * TDIM;  // 71.3M floats
  float* bX = alloc(NX);
  float* bY = alloc(NX);
  float* bXY = alloc(NX);
  float* bH = alloc(NX);
  float* bT = alloc(NT);
  float* bEF = alloc(NT);
  float* cbn = alloc(KCLS);
  int* idx = (int*)alloc((size_t)BN * TNUM);

  float* outj = (float*)d_out;
  float* loss = outj + (size_t)BN * JN * 3;

  const unsigned rows = BN * JN;  // 69632

  k_zero<<<1, 1, 0, stream>>>(loss);
  k_start<<<(unsigned)(NX / 256), 256, 0, stream>>>(joints, vis, F(2), F(3),
                                                    F(4), bX);
  // -------- encoder mixer blocks
  for (int blk = 0; blk < 4; ++blk) {
    const int p = 5 + blk * 12;
    k_ln512<<<rows / 8, 256, 0, stream>>>(bX, F(p + 0), F(p + 1), bY);
    k_tokmix<<<(BN * HIDD) / 256, 256, 0, stream>>>(bX, bY, F(p + 2), F(p + 3),
                                                    F(p + 4), F(p + 5), bXY);
    k_ln512<<<rows / 8, 256, 0, stream>>>(bXY, F(p + 6), F(p + 7), bY);
    dim3 g1(rows / 128, HIDD / 64);
    k_gemm_f32<<<g1, 256, 0, stream>>>(bY, F(p + 8), F(p + 9), nullptr, bH,
                                       rows, HIDD, HIDD, 1);
    k_gemm_f32<<<g1, 256, 0, stream>>>(bH, F(p + 10), F(p + 11), bXY, bX, rows,
                                       HIDD, HIDD, 0);
  }
  // -------- encoder head
  k_ln512<<<rows / 8, 256, 0, stream>>>(bX, F(53), F(54), bY);
  k_tokproj<<<(BN * HIDD) / 256, 256, 0, stream>>>(bY, F(55), F(56), bT);
  dim3 g2((BN * TNUM) / 128, TDIM / 64);
  k_gemm_f32<<<g2, 256, 0, stream>>>(bT, F(57), F(58), nullptr, bEF, BN * TNUM,
                                     TDIM, HIDD, 0);
  // -------- VQ nearest neighbor + latent loss
  k_rownorm<<<KCLS / 8, 256, 0, stream>>>(F(59), cbn);
  k_vq<<<(BN * TNUM) / 16, 128, 0, stream>>>(
      bEF, F(59), cbn, idx, loss,
      1.0f / ((float)BN * (float)TNUM * (float)TDIM));
  // -------- decoder
  k_dectok<<<(BN * TDIM) / 256, 256, 0, stream>>>(idx, F(59), F(60), F(61), bX);
  dim3 g3(rows / 128, 1);
  k_gemm_f32<<<g3, 256, 0, stream>>>(bX, F(62), F(63), nullptr, bY, rows, DHID,
                                     TDIM, 0);
  const int q = 64;  // dec[0] params
  k_dec<<<BN, 544, 0, stream>>>(bY, F(q + 0), F(q + 1), F(q + 2), F(q + 3),
                                F(q + 4), F(q + 5), F(q + 6), F(q + 7),
                                F(q + 8), F(q + 9), F(q + 10), F(q + 11),
                                F(76), F(77), F(78), F(79), outj);
}